// PGGCNModel_429496730127
// MI455X (gfx1250) — compile-verified
//
#include <hip/hip_runtime.h>

typedef __attribute__((ext_vector_type(2))) float v2f;
typedef __attribute__((ext_vector_type(8))) float v8f;

#define B_    1024
#define A_    200
#define F_    53
#define NAF_  38
#define ROUT_ 20
#define COUT_ 1024
#define ALDS_STRIDE 39   // 38 padded (odd stride -> conflict-free)
#define HLDS_STRIDE 21   // 20 padded
#define MAXROWS     208  // 13 M-tiles of 16

__launch_bounds__(256)
__global__ void pggcn_kernel(const float* __restrict__ inputs,
                             const int*   __restrict__ i_s,
                             const float* __restrict__ W_self,
                             const float* __restrict__ W_nei,
                             const float* __restrict__ b_r,
                             const float* __restrict__ Wc,
                             const float* __restrict__ bc,
                             const float* __restrict__ W1,
                             const float* __restrict__ b1,
                             const float* __restrict__ W5,
                             const float* __restrict__ b5,
                             const float* __restrict__ W6,
                             const float* __restrict__ b6,
                             const float* __restrict__ W7,
                             const float* __restrict__ b7,
                             float* __restrict__ out)
{
    __shared__ float atomLds[A_ * ALDS_STRIDE];      // 31200 B
    __shared__ float hLds[MAXROWS * HLDS_STRIDE];    // 17472 B
    __shared__ float wSelfLds[NAF_ * ROUT_];
    __shared__ float aggLds[NAF_];
    __shared__ float aggWLds[ROUT_];
    __shared__ float pooledLds[COUT_];
    __shared__ float partLds[256];
    __shared__ float d1Lds[32];
    __shared__ float d5Lds[16];
    __shared__ float physLds[15];

    const int b    = blockIdx.x;
    const int t    = threadIdx.x;
    const int lane = t & 31;
    const int wave = t >> 5;

    int nv = i_s[b];
    nv = nv < 1 ? 1 : (nv > A_ ? A_ : nv);
    const int nMt = (nv + 15) >> 4;

    const float* in_b = inputs + (size_t)b * (A_ * F_);

    // ---- Stage 0: cooperative loads + zero h padding ----
    for (int i = t; i < nv * NAF_; i += 256) {
        int a = i / NAF_, f = i - a * NAF_;
        atomLds[a * ALDS_STRIDE + f] = in_b[a * F_ + f];
    }
    for (int i = t; i < MAXROWS * HLDS_STRIDE; i += 256) hLds[i] = 0.f;
    for (int i = t; i < NAF_ * ROUT_; i += 256) wSelfLds[i] = W_self[i];
    if (t < 15) physLds[t] = in_b[NAF_ + t];
    __syncthreads();

    // ---- Stage 1: agg over valid atoms (38 features) ----
    if (t < NAF_) {
        float s = 0.f;
        for (int a = 0; a < nv; ++a) s += atomLds[a * ALDS_STRIDE + t];
        aggLds[t] = s;
    }
    __syncthreads();

    // aggW = agg @ W_nei  (20 outputs)
    if (t < ROUT_) {
        float s = 0.f;
        for (int f = 0; f < NAF_; ++f) s += aggLds[f] * W_nei[f * ROUT_ + t];
        aggWLds[t] = s;
    }
    __syncthreads();

    // ---- Stage 2: h = relu(atom@W_self + aggW + b_r), valid rows only ----
    // Padded rows (a >= nv) stay exactly 0 in hLds; their pooled contribution
    // is therefore exactly relu(bc[n]) each, corrected in closed form below.
    for (int e = t; e < nv * ROUT_; e += 256) {
        int a = e / ROUT_, r = e - a * ROUT_;
        float s = aggWLds[r] + b_r[r];
        for (int f = 0; f < NAF_; ++f)
            s += atomLds[a * ALDS_STRIDE + f] * wSelfLds[f * ROUT_ + r];
        hLds[a * HLDS_STRIDE + r] = fmaxf(s, 0.f);
    }
    __syncthreads();

    // ---- Stage 3: pooled = sum_a relu(h @ Wc + bc) via V_WMMA_F32_16X16X4_F32 ----
    // A frag (16x4 f32): lane L holds row (L&15), K-pair (L>=16 ? {2,3} : {0,1})
    // B frag (4x16 f32): lane L holds col (L&15), same K-pair split
    // C (16x16 f32): VGPR r = row r (lanes 0-15) / row r+8 (lanes 16-31), col lane&15
    const int mloc  = lane & 15;
    const int khalf = (lane >> 4) << 1;  // 0 or 2
    const float padRows = (float)(nMt * 16 - nv);
    for (int nt = wave * 8; nt < wave * 8 + 8; ++nt) {
        const int n0 = nt << 4;
        const int n  = n0 + mloc;
        v2f bf[5];
        #pragma unroll
        for (int kk = 0; kk < 5; ++kk) {
            int kbase = kk * 4 + khalf;
            bf[kk].x = Wc[(kbase + 0) * COUT_ + n];
            bf[kk].y = Wc[(kbase + 1) * COUT_ + n];
        }
        const float bcv = bc[n];
        float pa[8] = {0.f,0.f,0.f,0.f,0.f,0.f,0.f,0.f};
        const float* hrow = &hLds[mloc * HLDS_STRIDE + khalf];
        for (int mt = 0; mt < nMt; ++mt) {
            v8f c = {0.f,0.f,0.f,0.f,0.f,0.f,0.f,0.f};
            #pragma unroll
            for (int kk = 0; kk < 5; ++kk) {
                v2f af;
                af.x = hrow[kk * 4 + 0];
                af.y = hrow[kk * 4 + 1];
                c = __builtin_amdgcn_wmma_f32_16x16x4_f32(
                        false, af, false, bf[kk], (short)0, c, false, false);
            }
            #pragma unroll
            for (int r = 0; r < 8; ++r)
                pa[r] += fmaxf(c[r] + bcv, 0.f);
            hrow += 16 * HLDS_STRIDE;
        }
        float tot = pa[0]+pa[1]+pa[2]+pa[3]+pa[4]+pa[5]+pa[6]+pa[7];
        tot += __shfl_xor(tot, 16, 32);       // combine row halves (wave32)
        tot -= padRows * fmaxf(bcv, 0.f);     // exact removal of zero-row padding
        if (lane < 16) pooledLds[n0 + lane] = tot;
    }
    __syncthreads();

    // ---- Stage 4: MLP head ----
    // d1 = relu(pooled @ W1 + b1): 8 segments x 32 outputs
    {
        int r = t & 31, seg = t >> 5;
        float s = 0.f;
        int j0 = seg * 128;
        for (int j = j0; j < j0 + 128; ++j) s += pooledLds[j] * W1[j * 32 + r];
        partLds[t] = s;
    }
    __syncthreads();
    if (t < 32) {
        float s = b1[t];
        for (int seg = 0; seg < 8; ++seg) s += partLds[seg * 32 + t];
        d1Lds[t] = fmaxf(s, 0.f);
    }
    __syncthreads();
    if (t < 16) {
        float s = b5[t];
        for (int j = 0; j < 32; ++j) s += d1Lds[j] * W5[j * 16 + t];
        d5Lds[t] = fmaxf(s, 0.f);
    }
    __syncthreads();
    if (t == 0) {
        float mv = b6[0];
        for (int j = 0; j < 16; ++j) mv += d5Lds[j] * W6[j];
        float o = b7[0] + mv * W7[0];
        for (int i = 1; i < 16; ++i) o += physLds[i - 1] * W7[i];
        out[(size_t)b * 16] = o;
    }
    if (t >= 1 && t < 16) out[(size_t)b * 16 + t] = physLds[t - 1];
}

extern "C" void kernel_launch(void* const* d_in, const int* in_sizes, int n_in,
                              void* d_out, int out_size, void* d_ws, size_t ws_size,
                              hipStream_t stream) {
    const float* inputs = (const float*)d_in[0];
    const int*   i_s    = (const int*)  d_in[1];
    const float* W_self = (const float*)d_in[2];
    const float* W_nei  = (const float*)d_in[3];
    const float* b_r    = (const float*)d_in[4];
    const float* Wc     = (const float*)d_in[5];
    const float* bc     = (const float*)d_in[6];
    const float* W1     = (const float*)d_in[7];
    const float* b1     = (const float*)d_in[8];
    const float* W5     = (const float*)d_in[9];
    const float* b5     = (const float*)d_in[10];
    const float* W6     = (const float*)d_in[11];
    const float* b6     = (const float*)d_in[12];
    const float* W7     = (const float*)d_in[13];
    const float* b7     = (const float*)d_in[14];
    float* out = (float*)d_out;

    pggcn_kernel<<<B_, 256, 0, stream>>>(inputs, i_s, W_self, W_nei, b_r, Wc, bc,
                                         W1, b1, W5, b5, W6, b6, W7, b7, out);
}